// Graphsage_Encoder_6459630814178
// MI455X (gfx1250) — compile-verified
//
#include <hip/hip_runtime.h>

typedef float v2f __attribute__((ext_vector_type(2)));
typedef float v8f __attribute__((ext_vector_type(8)));

#define EPSF 1e-5f

// ---------------- zero a float region ----------------
__global__ void k_zero(float* __restrict__ p, int n) {
    int i = blockIdx.x * blockDim.x + threadIdx.x;
    if (i < n) p[i] = 0.f;
}

// ---------------- exclusive prefix scan of segment counts (tiny) ----------------
__global__ void k_scan(const int* __restrict__ counts, int* __restrict__ offs, int J) {
    if (threadIdx.x == 0 && blockIdx.x == 0) {
        int s = 0;
        for (int j = 0; j < J; ++j) { offs[j] = s; s += counts[j]; }
        offs[J] = s;
    }
}

// ---------------- fuse proj into the 4 first-layer weights ----------------
// Wf[k][n] = sum_j projW[k][j]*lW[j][n]  (k<256, j<128, n<256)
// bf[n]    = sum_j projb[j]*lW[j][n] + lb[n]
__global__ void k_fuse(const float* __restrict__ projW, const float* __restrict__ projb,
                       const float* __restrict__ lW0, const float* __restrict__ lb0,
                       const float* __restrict__ lW1, const float* __restrict__ lb1,
                       const float* __restrict__ lW2, const float* __restrict__ lb2,
                       const float* __restrict__ lW3, const float* __restrict__ lb3,
                       float* __restrict__ WF, float* __restrict__ BF) {
    int tid = blockIdx.x * blockDim.x + threadIdx.x;
    if (tid >= 4 * 256 * 256) return;
    int layer = tid >> 16;
    int rem   = tid & 0xFFFF;
    int k = rem >> 8;
    int n = rem & 255;
    const float* lW = layer == 0 ? lW0 : layer == 1 ? lW1 : layer == 2 ? lW2 : lW3;
    const float* lb = layer == 0 ? lb0 : layer == 1 ? lb1 : layer == 2 ? lb2 : lb3;
    float acc = 0.f;
    for (int j = 0; j < 128; ++j) acc += projW[k * 128 + j] * lW[j * 256 + n];
    WF[(size_t)layer * 65536 + k * 256 + n] = acc;
    if (k == 0) {
        float bacc = lb[n];
        for (int j = 0; j < 128; ++j) bacc += projb[j] * lW[j * 256 + n];
        BF[layer * 256 + n] = bacc;
    }
}

// ---------------- pack W[K,N] into k-pair-interleaved layout ----------------
// P[kp*N + n] = float2{ W[2kp][n], W[2kp+1][n] }  -> B fragment = one b64 load
__global__ void k_pack(const float* __restrict__ W, float* __restrict__ P, int N, int K) {
    int tid = blockIdx.x * blockDim.x + threadIdx.x;
    int total = (K >> 1) * N;
    if (tid >= total) return;
    int kp = tid / N;
    int n  = tid - kp * N;
    v2f v;
    v.x = W[(size_t)(2 * kp) * N + n];
    v.y = W[(size_t)(2 * kp + 1) * N + n];
    ((v2f*)P)[tid] = v;
}

// ---------------- WMMA f32 GEMM: Y = A[M,K] @ W[K,N] + bias, + BN group stats ----
// Wp is the k-pair packed weight (from k_pack). group(m,c) = ((m%Mb)/rowdiv)*rowmul + (c>>cshift).
// stats: [0..G)=sum, [G..2G)=sumsq (pre-zeroed). Block: 128 thr = 4 waves; wave tile
// 32(M) x 64(N); block tile 128 x 64. Full tiles run a software-pipelined loop.
__global__ void __launch_bounds__(128)
k_gemm(const float* __restrict__ A, const float* __restrict__ Wp,
       const float* __restrict__ bias, float* __restrict__ Y,
       float* __restrict__ stats, int M, int N, int K,
       int Mb, int rowdiv, int rowmul, int cshift, int nGroups) {
    __shared__ float s_sum[1024];
    __shared__ float s_sq[1024];

    const int lane = threadIdx.x & 31;
    const int wave = threadIdx.x >> 5;
    const int half = lane >> 4;   // 0 or 1
    const int l16  = lane & 15;
    const int row0 = blockIdx.y * 128 + wave * 32;
    const int col0 = blockIdx.x * 64;

    if (stats) {
        for (int i = threadIdx.x; i < nGroups; i += 128) { s_sum[i] = 0.f; s_sq[i] = 0.f; }
        __syncthreads();
    }

    const int  r0 = row0 + l16;
    const int  r1 = row0 + 16 + l16;
    const bool v0 = r0 < M, v1 = r1 < M;
    const float* aP0 = A + (size_t)(v0 ? r0 : 0) * K + 2 * half;
    const float* aP1 = A + (size_t)(v1 ? r1 : 0) * K + 2 * half;
    // packed B pointer: element (k0/2 + half)*N + col0 + l16, advance 2*N per k-step
    const v2f* wB = (const v2f*)Wp + (size_t)half * N + col0 + l16;

    const v8f vz = {0.f, 0.f, 0.f, 0.f, 0.f, 0.f, 0.f, 0.f};
    v8f acc[2][4];
#pragma unroll
    for (int mb = 0; mb < 2; ++mb)
#pragma unroll
        for (int t = 0; t < 4; ++t) acc[mb][t] = vz;

    if (row0 + 32 <= M) {
        // ---- full tile: software-pipelined, no guards ----
        v2f a0 = *(const v2f*)aP0;
        v2f a1 = *(const v2f*)aP1;
        v2f b0 = wB[0], b1 = wB[16], b2 = wB[32], b3 = wB[48];
        const int steps = (K >> 2) - 1;
        for (int s = 0; s < steps; ++s) {
            aP0 += 4; aP1 += 4; wB += (size_t)2 * N;
            v2f a0n = *(const v2f*)aP0;
            v2f a1n = *(const v2f*)aP1;
            v2f c0 = wB[0], c1 = wB[16], c2 = wB[32], c3 = wB[48];
            acc[0][0] = __builtin_amdgcn_wmma_f32_16x16x4_f32(false, a0, false, b0, (short)0, acc[0][0], false, false);
            acc[0][1] = __builtin_amdgcn_wmma_f32_16x16x4_f32(false, a0, false, b1, (short)0, acc[0][1], false, false);
            acc[0][2] = __builtin_amdgcn_wmma_f32_16x16x4_f32(false, a0, false, b2, (short)0, acc[0][2], false, false);
            acc[0][3] = __builtin_amdgcn_wmma_f32_16x16x4_f32(false, a0, false, b3, (short)0, acc[0][3], false, false);
            acc[1][0] = __builtin_amdgcn_wmma_f32_16x16x4_f32(false, a1, false, b0, (short)0, acc[1][0], false, false);
            acc[1][1] = __builtin_amdgcn_wmma_f32_16x16x4_f32(false, a1, false, b1, (short)0, acc[1][1], false, false);
            acc[1][2] = __builtin_amdgcn_wmma_f32_16x16x4_f32(false, a1, false, b2, (short)0, acc[1][2], false, false);
            acc[1][3] = __builtin_amdgcn_wmma_f32_16x16x4_f32(false, a1, false, b3, (short)0, acc[1][3], false, false);
            a0 = a0n; a1 = a1n; b0 = c0; b1 = c1; b2 = c2; b3 = c3;
        }
        acc[0][0] = __builtin_amdgcn_wmma_f32_16x16x4_f32(false, a0, false, b0, (short)0, acc[0][0], false, false);
        acc[0][1] = __builtin_amdgcn_wmma_f32_16x16x4_f32(false, a0, false, b1, (short)0, acc[0][1], false, false);
        acc[0][2] = __builtin_amdgcn_wmma_f32_16x16x4_f32(false, a0, false, b2, (short)0, acc[0][2], false, false);
        acc[0][3] = __builtin_amdgcn_wmma_f32_16x16x4_f32(false, a0, false, b3, (short)0, acc[0][3], false, false);
        acc[1][0] = __builtin_amdgcn_wmma_f32_16x16x4_f32(false, a1, false, b0, (short)0, acc[1][0], false, false);
        acc[1][1] = __builtin_amdgcn_wmma_f32_16x16x4_f32(false, a1, false, b1, (short)0, acc[1][1], false, false);
        acc[1][2] = __builtin_amdgcn_wmma_f32_16x16x4_f32(false, a1, false, b2, (short)0, acc[1][2], false, false);
        acc[1][3] = __builtin_amdgcn_wmma_f32_16x16x4_f32(false, a1, false, b3, (short)0, acc[1][3], false, false);
    } else {
        // ---- edge tile (only tiny M=8 GEMMs hit this): guarded, simple ----
        const v2f vz2 = {0.f, 0.f};
        for (int k0 = 0; k0 < K; k0 += 4) {
            v2f a0 = vz2, a1 = vz2;
            if (v0) a0 = *(const v2f*)(aP0 + k0);
            if (v1) a1 = *(const v2f*)(aP1 + k0);
            const v2f* w = wB + (size_t)(k0 >> 1) * N;
            v2f b0 = w[0], b1 = w[16], b2 = w[32], b3 = w[48];
            acc[0][0] = __builtin_amdgcn_wmma_f32_16x16x4_f32(false, a0, false, b0, (short)0, acc[0][0], false, false);
            acc[0][1] = __builtin_amdgcn_wmma_f32_16x16x4_f32(false, a0, false, b1, (short)0, acc[0][1], false, false);
            acc[0][2] = __builtin_amdgcn_wmma_f32_16x16x4_f32(false, a0, false, b2, (short)0, acc[0][2], false, false);
            acc[0][3] = __builtin_amdgcn_wmma_f32_16x16x4_f32(false, a0, false, b3, (short)0, acc[0][3], false, false);
            acc[1][0] = __builtin_amdgcn_wmma_f32_16x16x4_f32(false, a1, false, b0, (short)0, acc[1][0], false, false);
            acc[1][1] = __builtin_amdgcn_wmma_f32_16x16x4_f32(false, a1, false, b1, (short)0, acc[1][1], false, false);
            acc[1][2] = __builtin_amdgcn_wmma_f32_16x16x4_f32(false, a1, false, b2, (short)0, acc[1][2], false, false);
            acc[1][3] = __builtin_amdgcn_wmma_f32_16x16x4_f32(false, a1, false, b3, (short)0, acc[1][3], false, false);
        }
    }

    // Epilogue: bias, store, group stats (run-length folded, then LDS atomics).
    int prevG = -1; float ps = 0.f, pq = 0.f;
#pragma unroll
    for (int mb = 0; mb < 2; ++mb) {
#pragma unroll
        for (int g = 0; g < 8; ++g) {
            int m = row0 + 16 * mb + 8 * half + g;  // C/D layout: VGPR g = M row g (+8 hi half)
            if (m < M) {
#pragma unroll
                for (int t = 0; t < 4; ++t) {
                    int n = col0 + t * 16 + l16;
                    float v = acc[mb][t][g] + bias[n];
                    Y[(size_t)m * N + n] = v;
                    if (stats) {
                        int grp = ((m % Mb) / rowdiv) * rowmul + (n >> cshift);
                        if (grp != prevG) {
                            if (prevG >= 0) { atomicAdd(&s_sum[prevG], ps); atomicAdd(&s_sq[prevG], pq); }
                            prevG = grp; ps = 0.f; pq = 0.f;
                        }
                        ps += v; pq += v * v;
                    }
                }
            }
        }
    }
    if (stats) {
        if (prevG >= 0) { atomicAdd(&s_sum[prevG], ps); atomicAdd(&s_sq[prevG], pq); }
        __syncthreads();
        for (int i = threadIdx.x; i < nGroups; i += 128) {
            float s = s_sum[i], q = s_sq[i];
            if (s != 0.f || q != 0.f) {
                atomicAdd(&stats[i], s);
                atomicAdd(&stats[nGroups + i], q);
            }
        }
    }
}

// ---------------- BN (view-style groups) + optional relu + segment mean --------
__global__ void k_bn_seg(const float* __restrict__ Yin, const float* __restrict__ stats,
                         const float* __restrict__ gamma, const float* __restrict__ beta,
                         float* __restrict__ out, int N, int Mb,
                         int rowdiv, int rowmul, int cshift, int nGroups, float invCnt,
                         int relu, const int* __restrict__ offs, int J, int segLen,
                         int outStride, int colOff, int B) {
    int tid = blockIdx.x * blockDim.x + threadIdx.x;
    int total = B * J * N;
    if (tid >= total) return;
    int c  = tid % N;
    int ro = tid / N;
    int b  = ro / J;
    int j  = ro % J;
    int start, len;
    if (offs) { start = offs[j]; len = offs[j + 1] - start; }
    else      { start = j * segLen; len = segLen; }
    const float* rowp = Yin + ((size_t)b * Mb + start) * N + c;
    float accv = 0.f;
    for (int i = 0; i < len; ++i) {
        int m   = start + i;
        int grp = (m / rowdiv) * rowmul + (c >> cshift);
        float mu  = stats[grp] * invCnt;
        float var = stats[nGroups + grp] * invCnt - mu * mu;
        float scl = gamma[grp] * rsqrtf(var + EPSF);
        float v   = (rowp[(size_t)i * N] - mu) * scl + beta[grp];
        if (relu) v = fmaxf(v, 0.f);
        accv += v;
    }
    out[(size_t)(b * J + j) * outStride + colOff + c] = accv / (float)len;
}

// =============================== host launch ===================================
extern "C" void kernel_launch(void* const* d_in, const int* in_sizes, int n_in,
                              void* d_out, int out_size, void* d_ws, size_t ws_size,
                              hipStream_t stream) {
    (void)in_sizes; (void)n_in; (void)out_size; (void)ws_size;
    const float* x0 = (const float*)d_in[0];   // [8,32768,256]
    const float* x1 = (const float*)d_in[1];   // [8,2048,256]
    const float* x2 = (const float*)d_in[2];   // [8,64,256]
    const float* x3 = (const float*)d_in[3];   // [8,1,256]
    const int* counts0 = (const int*)d_in[4];  // [1,2048] (16s)
    const int* counts1 = (const int*)d_in[5];  // [1,64]   (32s)
    const float* projW = (const float*)d_in[6];  const float* projb = (const float*)d_in[7];
    const float* l30W  = (const float*)d_in[8];  const float* l30b  = (const float*)d_in[9];
    const float* l31W  = (const float*)d_in[10]; const float* l31b  = (const float*)d_in[11];
    const float* l32W  = (const float*)d_in[12]; const float* l32b  = (const float*)d_in[13];
    const float* l20W  = (const float*)d_in[14]; const float* l20b  = (const float*)d_in[15];
    const float* l21W  = (const float*)d_in[16]; const float* l21b  = (const float*)d_in[17];
    const float* l10W  = (const float*)d_in[18]; const float* l10b  = (const float*)d_in[19];
    const float* l00W  = (const float*)d_in[20]; const float* l00b  = (const float*)d_in[21];
    const float* loutW = (const float*)d_in[22]; const float* loutb = (const float*)d_in[23];
    const float* g30 = (const float*)d_in[24]; const float* be30 = (const float*)d_in[25];
    const float* g31 = (const float*)d_in[26]; const float* be31 = (const float*)d_in[27];
    const float* g32 = (const float*)d_in[28]; const float* be32 = (const float*)d_in[29];
    const float* g20 = (const float*)d_in[30]; const float* be20 = (const float*)d_in[31];
    const float* g21 = (const float*)d_in[32]; const float* be21 = (const float*)d_in[33];
    const float* g10 = (const float*)d_in[34]; const float* be10 = (const float*)d_in[35];
    const float* g00 = (const float*)d_in[36]; const float* be00 = (const float*)d_in[37];

    float* ws = (float*)d_ws;
    size_t cur = 0;
    auto alloc = [&](size_t nflt) { float* p = ws + cur; cur += nflt; return p; };

    float* WF   = alloc(4 * 65536);  // fused first-layer weights (row layout)
    float* BF   = alloc(4 * 256);
    float* st30 = alloc(512);        // stats regions contiguous: 6144 floats total
    float* st31 = alloc(1024);
    float* st32 = alloc(2048);
    float* st20 = alloc(512);
    float* st21 = alloc(1024);
    float* st10 = alloc(512);
    float* st00 = alloc(512);
    int*   off0 = (int*)alloc(2052);
    int*   off1 = (int*)alloc(68);
    float* PWF  = alloc(4 * 65536);              // packed fused weights
    float* P31  = alloc((size_t)256 * 512);      // packed l31W
    float* P32  = alloc((size_t)512 * 1024);     // packed l32W
    float* P21  = alloc((size_t)256 * 512);      // packed l21W
    float* PLO  = alloc((size_t)2048 * 2048);    // packed loutW
    float* Y0   = alloc((size_t)262144 * 256);
    float* S0   = alloc((size_t)16384 * 256);
    float* H1   = alloc((size_t)16384 * 512);
    float* S1   = alloc((size_t)512 * 512);
    float* H2   = alloc((size_t)512 * 1024);
    float* Y2   = alloc((size_t)16384 * 256);
    float* S2   = alloc((size_t)512 * 256);
    float* H21  = alloc((size_t)512 * 512);
    float* H10  = alloc((size_t)512 * 256);
    float* H00  = alloc((size_t)8 * 256);
    float* CC   = alloc((size_t)8 * 2048);

    float* Pf0 = PWF;             float* bf0 = BF;
    float* Pf1 = PWF + 65536;     float* bf1 = BF + 256;
    float* Pf2 = PWF + 2 * 65536; float* bf2 = BF + 512;
    float* Pf3 = PWF + 3 * 65536; float* bf3 = BF + 768;

    dim3 blk(128);
    auto grd = [](int M, int N) { return dim3((unsigned)(N / 64), (unsigned)((M + 127) / 128)); };

    // prep
    k_zero<<<(6144 + 255) / 256, 256, 0, stream>>>(st30, 6144);
    k_scan<<<1, 32, 0, stream>>>(counts0, off0, 2048);
    k_scan<<<1, 32, 0, stream>>>(counts1, off1, 64);
    k_fuse<<<(4 * 65536 + 255) / 256, 256, 0, stream>>>(
        projW, projb, l00W, l00b, l10W, l10b, l20W, l20b, l30W, l30b, WF, BF);
    // pack all GEMM weights into k-pair layout
    k_pack<<<(512 * 256 + 255) / 256, 256, 0, stream>>>(WF, PWF, 256, 1024);    // 4 fused mats
    k_pack<<<(128 * 512 + 255) / 256, 256, 0, stream>>>(l31W, P31, 512, 256);
    k_pack<<<(256 * 1024 + 255) / 256, 256, 0, stream>>>(l32W, P32, 1024, 512);
    k_pack<<<(128 * 512 + 255) / 256, 256, 0, stream>>>(l21W, P21, 512, 256);
    k_pack<<<(1024 * 2048 + 255) / 256, 256, 0, stream>>>(loutW, PLO, 2048, 2048);

    // ---- hop-3 branch ----
    k_gemm<<<grd(262144, 256), blk, 0, stream>>>(x0, Pf3, bf3, Y0, st30,
        262144, 256, 256, 32768, 128, 1, 31, 256);
    k_bn_seg<<<(8 * 2048 * 256 + 255) / 256, 256, 0, stream>>>(Y0, st30, g30, be30, S0,
        256, 32768, 128, 1, 31, 256, 1.f / 262144.f, 1, off0, 2048, 0, 256, 0, 8);
    k_gemm<<<grd(16384, 512), blk, 0, stream>>>(S0, P31, l31b, H1, st31,
        16384, 512, 256, 2048, 4, 1, 31, 512);
    k_bn_seg<<<(8 * 64 * 512 + 255) / 256, 256, 0, stream>>>(H1, st31, g31, be31, S1,
        512, 2048, 4, 1, 31, 512, 1.f / 16384.f, 1, off1, 64, 0, 512, 0, 8);
    k_gemm<<<grd(512, 1024), blk, 0, stream>>>(S1, P32, l32b, H2, st32,
        512, 1024, 512, 64, 1, 16, 6, 1024);
    k_bn_seg<<<(8 * 1024 + 255) / 256, 256, 0, stream>>>(H2, st32, g32, be32, CC,
        1024, 64, 1, 16, 6, 1024, 1.f / 512.f, 0, nullptr, 1, 64, 2048, 1024, 8);

    // ---- hop-2 branch ----
    k_gemm<<<grd(16384, 256), blk, 0, stream>>>(x1, Pf2, bf2, Y2, st20,
        16384, 256, 256, 2048, 8, 1, 31, 256);
    k_bn_seg<<<(8 * 64 * 256 + 255) / 256, 256, 0, stream>>>(Y2, st20, g20, be20, S2,
        256, 2048, 8, 1, 31, 256, 1.f / 16384.f, 1, off1, 64, 0, 256, 0, 8);
    k_gemm<<<grd(512, 512), blk, 0, stream>>>(S2, P21, l21b, H21, st21,
        512, 512, 256, 64, 1, 8, 6, 512);
    k_bn_seg<<<(8 * 512 + 255) / 256, 256, 0, stream>>>(H21, st21, g21, be21, CC,
        512, 64, 1, 8, 6, 512, 1.f / 512.f, 0, nullptr, 1, 64, 2048, 512, 8);

    // ---- hop-1 branch ----
    k_gemm<<<grd(512, 256), blk, 0, stream>>>(x2, Pf1, bf1, H10, st10,
        512, 256, 256, 64, 1, 4, 6, 256);
    k_bn_seg<<<(8 * 256 + 255) / 256, 256, 0, stream>>>(H10, st10, g10, be10, CC,
        256, 64, 1, 4, 6, 256, 1.f / 512.f, 0, nullptr, 1, 64, 2048, 256, 8);

    // ---- hop-0 branch ----
    k_gemm<<<grd(8, 256), blk, 0, stream>>>(x3, Pf0, bf0, H00, st00,
        8, 256, 256, 1, 1, 0, 0, 256);
    k_bn_seg<<<(8 * 256 + 255) / 256, 256, 0, stream>>>(H00, st00, g00, be00, CC,
        256, 1, 1, 0, 0, 256, 1.f / 8.f, 0, nullptr, 1, 1, 2048, 0, 8);

    // ---- output projection: out[8,2048] = CC @ loutW + loutb ----
    k_gemm<<<grd(8, 2048), blk, 0, stream>>>(CC, PLO, loutb, (float*)d_out, nullptr,
        8, 2048, 2048, 1, 1, 0, 0, 0);
}